// TGMandTRM_2482491097744
// MI455X (gfx1250) — compile-verified
//
#include <hip/hip_runtime.h>
#include <math.h>

#define C_CH 256
#define S_DIM 16
#define H_DIM 128
#define RANK 8

typedef __attribute__((ext_vector_type(2))) float v2f;
typedef __attribute__((ext_vector_type(4))) float v4f;
typedef __attribute__((ext_vector_type(8))) float v8f;

// Workspace layout (floats)
#define WS_SUMC 0                       // 256: sums leaving axis1 (c)
#define WS_SUMS 256                     // 16 : sums leaving axis2 (s)
#define WS_SUMH 272                     // 128: sums leaving axis4 (last dim)  -> Hv
#define WS_SUMW 400                     // 128: sums leaving axis3            -> Wv
#define WS_COEF 528                     // 1  : softmax(lam)[0]
#define WS_CF   544                     // 8*256
#define WS_SF   (WS_CF + RANK * C_CH)   // 8*16
#define WS_HF   (WS_SF + RANK * S_DIM)  // 8*128
#define WS_WF   (WS_HF + RANK * H_DIM)  // 8*128

// ---------------------------------------------------------------------------
// Pass 0: zero accumulators, compute coef = softmax(lam)[0]
// ---------------------------------------------------------------------------
__global__ void init_kernel(const float* __restrict__ lam, float* __restrict__ ws) {
    const int t = threadIdx.x;
    if (t < 528) ws[t] = 0.0f;
    if (t == 0) {
        float m = lam[0];
        #pragma unroll
        for (int i = 1; i < RANK; ++i) m = fmaxf(m, lam[i]);
        float den = 0.0f;
        #pragma unroll
        for (int i = 0; i < RANK; ++i) den += expf(lam[i] - m);
        ws[WS_COEF] = expf(lam[0] - m) / den;
    }
}

// ---------------------------------------------------------------------------
// Pass 1: four global-mean reductions in one streaming read of x.
// One wave handles one 128-element innermost row (v4f per lane).
// ---------------------------------------------------------------------------
__global__ void __launch_bounds__(256) reduce_kernel(const float* __restrict__ x,
                                                     float* __restrict__ ws) {
    __shared__ float sb[528];
    const int tid = threadIdx.x;
    for (int i = tid; i < 528; i += 256) sb[i] = 0.0f;
    __syncthreads();

    const int lane = tid & 31;
    const int wv   = tid >> 5;
    const int gw   = blockIdx.x * 8 + wv;
    const int nw   = gridDim.x * 8;
    const v4f* x4 = (const v4f*)x;

    for (int row = gw; row < C_CH * S_DIM * H_DIM; row += nw) {
        const v4f v = __builtin_nontemporal_load(&x4[(size_t)row * 32 + lane]);
        const int d3 = row & 127;
        const int cs = row >> 7;
        const int s  = cs & 15;
        const int c  = cs >> 4;
        // Hv bins (axis4 index = lane*4 + j): distinct bins per lane in a wave
        atomicAdd(&sb[WS_SUMH + lane * 4 + 0], v.x);
        atomicAdd(&sb[WS_SUMH + lane * 4 + 1], v.y);
        atomicAdd(&sb[WS_SUMH + lane * 4 + 2], v.z);
        atomicAdd(&sb[WS_SUMH + lane * 4 + 3], v.w);
        // row sum -> Cv, Sv, Wv bins
        float rs = v.x + v.y + v.z + v.w;
        #pragma unroll
        for (int off = 16; off > 0; off >>= 1) rs += __shfl_xor(rs, off);
        if (lane == 0) {
            atomicAdd(&sb[WS_SUMC + c],  rs);
            atomicAdd(&sb[WS_SUMS + s],  rs);
            atomicAdd(&sb[WS_SUMW + d3], rs);
        }
    }
    __syncthreads();
    for (int i = tid; i < 528; i += 256) atomicAdd(&ws[i], sb[i]);
}

// ---------------------------------------------------------------------------
// Pass 2: batched GEMV + sigmoid via V_WMMA_F32_16X16X4_F32.
// One wave per (rank, 16-wide output tile).
// A (16x4 f32 layout): lane0 holds row0 K0/K1, lane16 holds row0 K2/K3, all
// other lanes zero. Achieved branchlessly: every lane loads v2f from
// vsum+kh2+k0 and scales by msk = (lane==0 || lane==16) ? inv : 0.
// B (4x16): lanes 0-15 = K0/K1 of column `col`, lanes 16-31 = K2/K3.
// D row M=0 lands in acc[0] lanes 0-15 -> sigmoid -> store.
// ---------------------------------------------------------------------------
__global__ void __launch_bounds__(32) gemv_sigmoid_wmma(const float* __restrict__ W1,
                                                        const float* __restrict__ W2,
                                                        const float* __restrict__ W3,
                                                        const float* __restrict__ W4,
                                                        float* __restrict__ ws) {
    const int job  = blockIdx.x;   // 0..263
    const int lane = threadIdx.x;  // 0..31

    const float* W; const float* vsum; float* outp;
    int Nin, c0; float inv;
    if (job < 128) {                       // cf = sigmoid(W1 @ Cv)
        const int r = job >> 4; c0 = (job & 15) << 4;
        W = W1 + (size_t)r * C_CH * C_CH; Nin = C_CH;
        vsum = ws + WS_SUMC; inv = 1.0f / (S_DIM * H_DIM * H_DIM);
        outp = ws + WS_CF + r * C_CH;
    } else if (job < 192) {                // hf = sigmoid(W3 @ Hv)
        const int j = job - 128; const int r = j >> 3; c0 = (j & 7) << 4;
        W = W3 + (size_t)r * H_DIM * H_DIM; Nin = H_DIM;
        vsum = ws + WS_SUMH; inv = 1.0f / (C_CH * S_DIM * H_DIM);
        outp = ws + WS_HF + r * H_DIM;
    } else if (job < 256) {                // wf = sigmoid(W4 @ Wv)
        const int j = job - 192; const int r = j >> 3; c0 = (j & 7) << 4;
        W = W4 + (size_t)r * H_DIM * H_DIM; Nin = H_DIM;
        vsum = ws + WS_SUMW; inv = 1.0f / (C_CH * S_DIM * H_DIM);
        outp = ws + WS_WF + r * H_DIM;
    } else {                               // sf = sigmoid(W2 @ Sv)
        const int r = job - 256; c0 = 0;
        W = W2 + (size_t)r * S_DIM * S_DIM; Nin = S_DIM;
        vsum = ws + WS_SUMS; inv = 1.0f / (C_CH * H_DIM * H_DIM);
        outp = ws + WS_SF + r * S_DIM;
    }

    const int col = lane & 15;
    const int kh2 = (lane >> 4) << 1;       // 0 for lanes 0-15, 2 for lanes 16-31
    // Per-lane A mask: only lanes 0 and 16 carry row 0 of A; fold in the mean
    // divisor. All other lanes contribute zeros (rows 1..15 of A -> ignored).
    const float msk  = (lane == 0 || lane == 16) ? inv : 0.0f;
    const float* ap   = vsum + kh2;         // lane-dependent A source
    const float* brow = W + (size_t)(c0 + col) * Nin + kh2;

    v8f acc = {0.f, 0.f, 0.f, 0.f, 0.f, 0.f, 0.f, 0.f};

    #pragma unroll 4
    for (int k0 = 0; k0 < Nin; k0 += 4) {
        const v2f av = *(const v2f*)(ap + k0);
        v2f a;
        a[0] = msk * av[0];
        a[1] = msk * av[1];
        const v2f b = *(const v2f*)(brow + k0);
        acc = __builtin_amdgcn_wmma_f32_16x16x4_f32(false, a, false, b,
                                                    (short)0, acc, false, false);
    }
    if (lane < 16) {
        const float v = acc[0];  // D[M=0][N=lane]
        outp[c0 + lane] = 1.0f / (1.0f + expf(-v));
    }
}

// ---------------------------------------------------------------------------
// Pass 3: streaming apply. One block per (c,s) slice of 128x128.
// out[:256ch] = x ; out[256ch:] = relu(x * t1), t1 = coef * sum_r cf sf hf wf.
// NT hints: x is read once more and out is write-once -> keep L2 clean.
// ---------------------------------------------------------------------------
__global__ void __launch_bounds__(256) apply_kernel(const float* __restrict__ x,
                                                    const float* __restrict__ ws,
                                                    float* __restrict__ out) {
    __shared__ float sh[RANK * H_DIM];  // hf (multiplies axis-3 index)
    __shared__ float sw[RANK * H_DIM];  // wf (multiplies axis-4 index)
    const int b   = blockIdx.x;  // 0..4095
    const int c   = b >> 4;
    const int s   = b & 15;
    const int tid = threadIdx.x;

    for (int i = tid; i < RANK * H_DIM; i += 256) {
        sh[i] = ws[WS_HF + i];
        sw[i] = ws[WS_WF + i];
    }
    __syncthreads();

    const float coef = ws[WS_COEF];
    float q[RANK];
    #pragma unroll
    for (int r = 0; r < RANK; ++r)
        q[r] = coef * ws[WS_CF + r * C_CH + c] * ws[WS_SF + r * S_DIM + s];

    const size_t base = (size_t)b * (H_DIM * H_DIM);
    const v4f* xin = (const v4f*)(x + base);
    v4f* o1 = (v4f*)(out + base);
    v4f* o2 = (v4f*)(out + base + (size_t)C_CH * S_DIM * H_DIM * H_DIM);

    for (int g = tid; g < (H_DIM * H_DIM) / 4; g += 256) {
        const int idx = g * 4;
        const int d3 = idx >> 7;
        const int d4 = idx & 127;
        const v4f v = __builtin_nontemporal_load(&xin[g]);
        float pr[RANK];
        #pragma unroll
        for (int r = 0; r < RANK; ++r) pr[r] = q[r] * sh[r * H_DIM + d3];
        float t0 = 0.f, t1 = 0.f, t2 = 0.f, t3 = 0.f;
        #pragma unroll
        for (int r = 0; r < RANK; ++r) {
            const float* wr = &sw[r * H_DIM + d4];
            t0 = fmaf(pr[r], wr[0], t0);
            t1 = fmaf(pr[r], wr[1], t1);
            t2 = fmaf(pr[r], wr[2], t2);
            t3 = fmaf(pr[r], wr[3], t3);
        }
        v4f y;
        y.x = fmaxf(v.x * t0, 0.f);
        y.y = fmaxf(v.y * t1, 0.f);
        y.z = fmaxf(v.z * t2, 0.f);
        y.w = fmaxf(v.w * t3, 0.f);
        __builtin_nontemporal_store(v, &o1[g]);  // concat part 1: copy of x
        __builtin_nontemporal_store(y, &o2[g]);  // concat part 2: relu(x * t1)
    }
}

extern "C" void kernel_launch(void* const* d_in, const int* in_sizes, int n_in,
                              void* d_out, int out_size, void* d_ws, size_t ws_size,
                              hipStream_t stream) {
    const float* x   = (const float*)d_in[0];
    const float* W1  = (const float*)d_in[1];
    const float* W2  = (const float*)d_in[2];
    const float* W3  = (const float*)d_in[3];
    const float* W4  = (const float*)d_in[4];
    const float* lam = (const float*)d_in[5];
    float* out = (float*)d_out;
    float* ws  = (float*)d_ws;

    init_kernel<<<1, 544, 0, stream>>>(lam, ws);
    reduce_kernel<<<2048, 256, 0, stream>>>(x, ws);
    gemv_sigmoid_wmma<<<264, 32, 0, stream>>>(W1, W2, W3, W4, ws);
    apply_kernel<<<4096, 256, 0, stream>>>(x, ws, out);
}